// F_Loss_82952998355608
// MI455X (gfx1250) — compile-verified
//
#include <hip/hip_runtime.h>
#include <math.h>

// Problem constants (from reference)
#define B_IMG 8
#define A_N   49104
#define C_N   80
#define M_N   32
#define N_N   16
#define IMG_SZ 512.0f

#define TA 256                 // threads, IoU kernel
#define TB 512                 // threads, selection kernel
#define BMW ((A_N + 31) / 32)  // bitmap words per image (1535)

typedef float v2f __attribute__((ext_vector_type(2)));
typedef float v8f __attribute__((ext_vector_type(8)));

// ---------------------------------------------------------------------------
// gfx1250 async Global->LDS staging (ASYNCcnt-tracked DMA). Guarded so the
// kernel still compiles if the toolchain lacks the async builtins.
// ---------------------------------------------------------------------------
#if __has_builtin(__builtin_amdgcn_global_load_async_to_lds_b32)
#define HAVE_ASYNC_LDS 1
#if __has_builtin(__builtin_amdgcn_s_wait_asynccnt)
#define ASYNC_WAIT() __builtin_amdgcn_s_wait_asynccnt(0)
#else
#define ASYNC_WAIT() asm volatile("s_wait_asynccnt 0x0" ::: "memory")
#endif
#else
#define HAVE_ASYNC_LDS 0
#define ASYNC_WAIT()
#endif

// Stage the 32x5 annotation rows for image b into LDS (160 dwords).
// Call with the whole block; follow with __syncthreads() at the call site.
__device__ __forceinline__ void stage_ann(const float* __restrict__ ann_img,
                                          float* s_ann, int tid) {
#if HAVE_ASYNC_LDS
  if (tid < M_N * 5) {
    __builtin_amdgcn_global_load_async_to_lds_b32(
        (__attribute__((address_space(1))) int*)(ann_img + tid),
        (__attribute__((address_space(3))) int*)(&s_ann[tid]),
        /*offset=*/0, /*cpol=*/0);
  }
  ASYNC_WAIT();
#else
  if (tid < M_N * 5) s_ann[tid] = ann_img[tid];
#endif
}

// ---------------------------------------------------------------------------
// WMMA-based sum of 64 f32 values held in LDS. A = ones(16x4), B = values
// (4x16, two per lane), D[i][j] = sum_k B[k,j]. Row 0 of D (VGPR0, lanes
// 0..15) holds the 16 column sums; a 16-wide xor butterfly finishes the
// reduction. Must be called by a full wave (EXEC all ones). Valid on lane 0.
// ---------------------------------------------------------------------------
__device__ __forceinline__ float wmma_sum64(const float* vals, int lane) {
  v2f a;  a.x = 1.0f;           a.y = 1.0f;
  v2f bv; bv.x = vals[2 * lane]; bv.y = vals[2 * lane + 1];
  v8f c = {0.f, 0.f, 0.f, 0.f, 0.f, 0.f, 0.f, 0.f};
  v8f d = __builtin_amdgcn_wmma_f32_16x16x4_f32(
      /*neg_a=*/false, a, /*neg_b=*/false, bv,
      /*c_mod=*/(short)0, c, /*reuse_a=*/false, /*reuse_b=*/false);
  float t = d[0];
  t += __shfl_xor(t, 1, 16);
  t += __shfl_xor(t, 2, 16);
  t += __shfl_xor(t, 4, 16);
  t += __shfl_xor(t, 8, 16);
  return t;
}

// ---------------------------------------------------------------------------
// Kernel 0: init per-image global-max bits (IoU >= 0 so 0u is identity).
// ---------------------------------------------------------------------------
__global__ void k_init(unsigned* gmax_bits) {
  int t = threadIdx.x;
  if (t < B_IMG) gmax_bits[t] = 0u;
}

// ---------------------------------------------------------------------------
// Kernel 1: per-anchor IoU max/argmax vs the 32 (clamped) GT boxes, plus a
// per-image global max of iou_max via monotone bit-pattern atomicMax.
// ---------------------------------------------------------------------------
__global__ void k_iou(const float* __restrict__ anchors,
                      const float* __restrict__ ann,
                      float* __restrict__ iou_max_ws,
                      int*   __restrict__ iou_arg_ws,
                      unsigned* __restrict__ gmax_bits) {
  const int b   = blockIdx.y;
  const int tid = threadIdx.x;
  const int a   = blockIdx.x * TA + tid;

  __shared__ float s_ann[M_N * 5];
  __shared__ float bx1[M_N], by1[M_N], bx2[M_N], by2[M_N], barea[M_N];

  stage_ann(ann + (size_t)b * M_N * 5, s_ann, tid);
  __syncthreads();
  if (tid < M_N) {
    const float* an = s_ann + tid * 5;
    float x1 = fminf(fmaxf(an[0], 0.f), IMG_SZ);
    float y1 = fminf(fmaxf(an[1], 0.f), IMG_SZ);
    float x2 = fminf(fmaxf(an[2], 0.f), IMG_SZ);
    float y2 = fminf(fmaxf(an[3], 0.f), IMG_SZ);
    bx1[tid] = x1; by1[tid] = y1; bx2[tid] = x2; by2[tid] = y2;
    barea[tid] = (x2 - x1) * (y2 - y1);
  }
  __syncthreads();

  float imax = 0.0f;
  if (a < A_N) {
    const float4 av = *reinterpret_cast<const float4*>(
        anchors + ((size_t)b * A_N + a) * 4);
    const float aw = av.z - av.x, ah = av.w - av.y;
    const float area_a = aw * ah;
    float best = -1.0f;
    int   barg = 0;
#pragma unroll 8
    for (int m = 0; m < M_N; ++m) {
      float iw = fmaxf(fminf(av.z, bx2[m]) - fmaxf(av.x, bx1[m]), 0.f);
      float ih = fmaxf(fminf(av.w, by2[m]) - fmaxf(av.y, by1[m]), 0.f);
      float inter = iw * ih;
      float ua    = fmaxf(area_a + barea[m] - inter, 1e-8f);
      float iou   = inter / ua;
      if (iou > best) { best = iou; barg = m; }   // first max wins (argmax)
    }
    iou_max_ws[(size_t)b * A_N + a] = best;
    iou_arg_ws[(size_t)b * A_N + a] = barg;
    imax = best;
  }

  __shared__ float red[TA];
  red[tid] = imax;
  __syncthreads();
  for (int off = TA / 2; off > 0; off >>= 1) {
    if (tid < off) red[tid] = fmaxf(red[tid], red[tid + off]);
    __syncthreads();
  }
  if (tid == 0) atomicMax(&gmax_bits[b], __float_as_uint(red[0]));
}

// ---------------------------------------------------------------------------
// Kernel 2: one block per image. Candidacy is frozen into LDS bitmaps, then
// exact deterministic top-k selection (iterative argmax over bitmap words,
// ties -> lowest index == argsort stability), then the cls / reg loss sums
// reduced with V_WMMA_F32_16X16X4_F32.
// ---------------------------------------------------------------------------
__global__ void __launch_bounds__(TB)
k_select(const float* __restrict__ cls,
         const float* __restrict__ regs,
         const float* __restrict__ anchors,
         const float* __restrict__ ann,
         const float* __restrict__ scores,
         const float* __restrict__ iou_max_ws,
         const int*   __restrict__ iou_arg_ws,
         const unsigned* __restrict__ gmax_bits,
         float* __restrict__ img_cls,
         float* __restrict__ img_reg) {
  const int b   = blockIdx.x;
  const int tid = threadIdx.x;
  const size_t base = (size_t)b * A_N;

  __shared__ unsigned bm_cand_pos[BMW];  // candidacy (immutable after build)
  __shared__ unsigned bm_cand_neg[BMW];
  __shared__ unsigned bm_pos[BMW];       // selected
  __shared__ unsigned bm_neg[BMW];
  __shared__ float s_sc[TB];
  __shared__ int   s_ix[TB];
  __shared__ float s_ann[M_N * 5];
  __shared__ float bx1[M_N], by1[M_N], bx2[M_N], by2[M_N];
  __shared__ int   blab[M_N];
  __shared__ int   sel_idx[4 * N_N];
  __shared__ int   sel_flag[4 * N_N];   // 1=pos, 2=neg, 0=neg shadowed by pos
  __shared__ int   s_np, s_nn;
  __shared__ float clsval[4 * N_N], regval[4 * N_N];

  stage_ann(ann + (size_t)b * M_N * 5, s_ann, tid);
  __builtin_prefetch(&scores[(base + tid) * 2], 0, 1);  // warm score stream

  for (int i = tid; i < BMW; i += TB) { bm_pos[i] = 0u; bm_neg[i] = 0u; }
  if (tid == 0) { s_np = 0; s_nn = 0; }
  if (tid < 4 * N_N) { sel_idx[tid] = 0; sel_flag[tid] = 0; }
  __syncthreads();
  if (tid < M_N) {
    const float* an = s_ann + tid * 5;
    bx1[tid] = fminf(fmaxf(an[0], 0.f), IMG_SZ);
    by1[tid] = fminf(fmaxf(an[1], 0.f), IMG_SZ);
    bx2[tid] = fminf(fmaxf(an[2], 0.f), IMG_SZ);
    by2[tid] = fminf(fmaxf(an[3], 0.f), IMG_SZ);
    blab[tid] = (int)an[4];
  }

  const float thresh = fminf(__uint_as_float(gmax_bits[b]), 0.7f);

  // build candidacy bitmaps + counts (one global pass over iou_max)
  int np = 0, nn = 0;
  for (int w = tid; w < BMW; w += TB) {
    unsigned cp = 0u, cn = 0u;
    const int abase = w << 5;
    const int lim = min(32, A_N - abase);
    for (int i = 0; i < lim; ++i) {
      float v = iou_max_ws[base + abase + i];
      if (v >= thresh) cp |= 1u << i;
      if (v >= 0.1f && v < 0.4f) cn |= 1u << i;
    }
    bm_cand_pos[w] = cp;
    bm_cand_neg[w] = cn;
    np += __popc(cp);
    nn += __popc(cn);
  }
  atomicAdd(&s_np, np);
  atomicAdd(&s_nn, nn);
  __syncthreads();

  const int k_pos = min(N_N, s_np);
  const int k_neg = min(4 * N_N - k_pos, s_nn);
  const int ksel  = k_pos + k_neg;

  // iterative argmax selection: pos passes first, then neg passes.
  for (int pass = 0; pass < ksel; ++pass) {
    const bool is_pos = (pass < k_pos);
    const unsigned* cand_bm = is_pos ? bm_cand_pos : bm_cand_neg;
    const unsigned* sel_bm  = is_pos ? bm_pos : bm_neg;
    float best_s = -1.0f;
    int   best_i = -1;
    for (int w = tid; w < BMW; w += TB) {
      unsigned m = cand_bm[w] & ~sel_bm[w];
      while (m) {
        int bit = __ffs(m) - 1;
        m &= m - 1;
        int a = (w << 5) + bit;
        float s = scores[(base + a) * 2 + (is_pos ? 0 : 1)];
        if (s > best_s) { best_s = s; best_i = a; }  // ascending a: first max
      }
    }
    s_sc[tid] = best_s;
    s_ix[tid] = best_i;
    __syncthreads();
    for (int off = TB / 2; off > 0; off >>= 1) {
      if (tid < off) {
        float so = s_sc[tid + off];
        int   io = s_ix[tid + off];
        if (so > s_sc[tid] ||
            (so == s_sc[tid] && io >= 0 && (s_ix[tid] < 0 || io < s_ix[tid]))) {
          s_sc[tid] = so; s_ix[tid] = io;
        }
      }
      __syncthreads();
    }
    if (tid == 0) {
      int wi = s_ix[0];
      if (wi >= 0) {
        if (is_pos) {
          bm_pos[wi >> 5] |= 1u << (wi & 31);
          sel_idx[pass] = wi;
          sel_flag[pass] = 1;
        } else {
          bm_neg[wi >> 5] |= 1u << (wi & 31);
          sel_idx[pass] = wi;
          // positives override negatives in the target assignment
          int shadowed = 0;
          for (int j = 0; j < k_pos; ++j)
            if (sel_idx[j] == wi) { shadowed = 1; break; }
          sel_flag[pass] = shadowed ? 0 : 2;
        }
      }
    }
    __syncthreads();
  }

  // per-slot loss terms (64 cls slots; 16 pos anchors x 4 coords reg slots)
  if (tid < 4 * N_N) {
    float v = 0.0f;
    if (tid < ksel && sel_flag[tid] != 0) {
      int a = sel_idx[tid];
      int cidx = (sel_flag[tid] == 1) ? blab[iou_arg_ws[base + a]] : (C_N - 1);
      float pr = cls[(base + a) * C_N + cidx];
      pr = fminf(fmaxf(pr, 1e-7f), 1.0f - 1e-7f);
      v = -logf(pr);
    }
    clsval[tid] = v;

    float rv = 0.0f;
    const int slot  = tid >> 2;
    const int coord = tid & 3;
    if (slot < k_pos) {
      int a = sel_idx[slot];
      const float4 av = *reinterpret_cast<const float4*>(
          anchors + (base + a) * 4);
      float aw = av.z - av.x, ah = av.w - av.y;
      float acx = av.x + 0.5f * aw, acy = av.y + 0.5f * ah;
      int m = iou_arg_ws[base + a];
      float gw = bx2[m] - bx1[m];
      float gh = by2[m] - by1[m];
      float gcx = bx1[m] + 0.5f * gw;   // centers use unclamped wh
      float gcy = by1[m] + 0.5f * gh;
      gw = fmaxf(gw, 1.0f);
      gh = fmaxf(gh, 1.0f);
      float rt;
      if      (coord == 0) rt = ((gcx - acx) / aw) * 10.0f;
      else if (coord == 1) rt = ((gcy - acy) / ah) * 10.0f;
      else if (coord == 2) rt = logf(gw / aw) * 5.0f;
      else                 rt = logf(gh / ah) * 5.0f;
      float rg = regs[(base + a) * 4 + coord];
      float diff = fabsf(rt - rg);
      rv = (diff <= (1.0f / 9.0f)) ? 4.5f * diff * diff
                                   : diff - (0.5f / 9.0f);
    }
    regval[tid] = rv;
  }
  __syncthreads();

  // wave 0 (full EXEC): matrix-pipe reduction of the two 64-value sets
  if (tid < warpSize) {
    float ct   = wmma_sum64(clsval, tid);
    float rtot = wmma_sum64(regval, tid);
    if (tid == 0) {
      img_cls[b] = ct   * (1.0f / (4.0f * N_N));
      img_reg[b] = rtot * (1.0f / (4.0f * N_N));
    }
  }
}

// ---------------------------------------------------------------------------
// Kernel 3: deterministic batch mean -> d_out[0]=cls, d_out[1]=reg.
// ---------------------------------------------------------------------------
__global__ void k_final(const float* __restrict__ img_cls,
                        const float* __restrict__ img_reg,
                        float* __restrict__ out) {
  if (threadIdx.x == 0) {
    float c = 0.f, r = 0.f;
    for (int i = 0; i < B_IMG; ++i) { c += img_cls[i]; r += img_reg[i]; }
    out[0] = c * (1.0f / B_IMG);
    out[1] = r * (1.0f / B_IMG);
  }
}

// ---------------------------------------------------------------------------
extern "C" void kernel_launch(void* const* d_in, const int* in_sizes, int n_in,
                              void* d_out, int out_size, void* d_ws, size_t ws_size,
                              hipStream_t stream) {
  const float* classifications = (const float*)d_in[0];
  const float* regressions     = (const float*)d_in[1];
  const float* anchors         = (const float*)d_in[2];
  const float* annotations     = (const float*)d_in[3];
  const float* sample_scores   = (const float*)d_in[4];
  float* out = (float*)d_out;

  // workspace layout
  float*    iou_max_ws = (float*)d_ws;
  int*      iou_arg_ws = (int*)(iou_max_ws + (size_t)B_IMG * A_N);
  unsigned* gmax_bits  = (unsigned*)(iou_arg_ws + (size_t)B_IMG * A_N);
  float*    img_cls    = (float*)(gmax_bits + B_IMG);
  float*    img_reg    = img_cls + B_IMG;

  k_init<<<1, 32, 0, stream>>>(gmax_bits);

  dim3 g1((A_N + TA - 1) / TA, B_IMG);
  k_iou<<<g1, TA, 0, stream>>>(anchors, annotations,
                               iou_max_ws, iou_arg_ws, gmax_bits);

  k_select<<<B_IMG, TB, 0, stream>>>(classifications, regressions, anchors,
                                     annotations, sample_scores,
                                     iou_max_ws, iou_arg_ws, gmax_bits,
                                     img_cls, img_reg);

  k_final<<<1, 32, 0, stream>>>(img_cls, img_reg, out);
}